// Dilate_35708358099161
// MI455X (gfx1250) — compile-verified
//
#include <hip/hip_runtime.h>

// Dilate (zero-insertion upsample), stride (2,2):
//   x: (16, 64, 256, 256) f32  ->  out: (16, 64, 512, 512) f32
//   out[b,c,2h,2w] = x[b,c,h,w]; all other outputs are 0.
//
// Pure data movement: 256 MB in + 1 GB out => ~54 us floor at 23.3 TB/s HBM.
// No FLOPs => WMMA is inapplicable; the optimization target is peak streaming
// bandwidth via b128 lane-contiguous stores with non-temporal (TH_NT) hints.

typedef float f32x2 __attribute__((ext_vector_type(2)));
typedef float f32x4 __attribute__((ext_vector_type(4)));

#define BC_PLANES (16 * 64)                         // 1024 (b,c) planes
#define H_IN 256
#define W_IN 256
#define THREADS_PER_ROW (W_IN / 2)                  // 128 float2's per input row
#define N_THREADS (BC_PLANES * H_IN * THREADS_PER_ROW)  // 33,554,432

__global__ __launch_bounds__(256) void dilate2x2_kernel(
    const float* __restrict__ in, float* __restrict__ out) {
  const int tid = blockIdx.x * blockDim.x + threadIdx.x;
  if (tid >= N_THREADS) return;

  const int wi  = tid & (THREADS_PER_ROW - 1);  // which float2 within input row
  const int row = tid >> 7;                     // = (b*64 + c)*256 + h
  const int h   = row & (H_IN - 1);
  const int bc  = row >> 8;

  // Load 2 consecutive input floats (global_load_b64, NT: read-once stream).
  const f32x2* in2 = reinterpret_cast<const f32x2*>(in);
  const f32x2 v = __builtin_nontemporal_load(&in2[tid]);

  const f32x4 even = {v.x, 0.0f, v.y, 0.0f};
  const f32x4 zero = {0.0f, 0.0f, 0.0f, 0.0f};

  // Output plane = 512*512 floats = 65536 float4; output row 2h starts at
  // float4 index h*256 within the plane; this thread's column chunk is +wi.
  // Consecutive lanes -> consecutive float4's: each wave32 b128 store covers
  // 512 contiguous bytes (full lines, no partial-line writes under NT).
  f32x4* out4 = reinterpret_cast<f32x4*>(out);
  const int ebase = bc * 65536 + h * 256 + wi;
  __builtin_nontemporal_store(even, &out4[ebase]);         // even row: data+0s
  __builtin_nontemporal_store(zero, &out4[ebase + 128]);   // odd row (+512 f32): zeros
}

extern "C" void kernel_launch(void* const* d_in, const int* in_sizes, int n_in,
                              void* d_out, int out_size, void* d_ws, size_t ws_size,
                              hipStream_t stream) {
  const float* x = (const float*)d_in[0];
  float* out = (float*)d_out;

  const int block = 256;                 // 8 wave32s per block
  const int grid = N_THREADS / block;    // 131,072 blocks
  dilate2x2_kernel<<<grid, block, 0, stream>>>(x, out);
}